// GCN_51445118271702
// MI455X (gfx1250) — compile-verified
//
#include <hip/hip_runtime.h>
#include <math.h>

typedef __attribute__((ext_vector_type(2))) float v2f;
typedef __attribute__((ext_vector_type(8))) float v8f;

#define N_NODES 100000
#define N_EDGES 3200000
#define D_IN    128
#define D_HID   16
#define N_CLS   10
#define N_TILES (N_NODES / 16)   // 6250, exact

// ---------------- degree / normalization ----------------

__global__ void k_init_deg(float* __restrict__ deg) {
    int n = blockIdx.x * blockDim.x + threadIdx.x;
    if (n < N_NODES) deg[n] = 1.0f;  // self-loop
}

__global__ void k_count_deg(const int* __restrict__ dst, float* __restrict__ deg) {
    int e = blockIdx.x * blockDim.x + threadIdx.x;
    if (e < N_EDGES) atomicAdd(&deg[dst[e]], 1.0f);
}

__global__ void k_deg_to_dinv(float* __restrict__ deg) {
    int n = blockIdx.x * blockDim.x + threadIdx.x;
    if (n < N_NODES) deg[n] = rsqrtf(deg[n]);  // deg >= 1 guaranteed
}

// ---------------- layer 1 GEMM: h1 = x @ W1 (WMMA f32 16x16x4) ----------------
// One wave computes one 16x16 output tile (16 rows of nodes, all 16 hidden dims).
// A layout (16x4 f32): lanes 0-15 -> M=lane, K={0,1}; lanes 16-31 -> M=lane-16, K={2,3}
// B layout (4x16 f32): lanes 0-15 -> N=lane, K={0,1}; lanes 16-31 -> N=lane-16, K={2,3}
// C/D layout: vgpr j, lanes 0-15 -> M=j, N=lane; lanes 16-31 -> M=j+8, N=lane-16

__global__ void k_gemm1(const float* __restrict__ x,
                        const float* __restrict__ W1,
                        float* __restrict__ h1) {
    const int wave = (blockIdx.x * blockDim.x + threadIdx.x) >> 5;
    if (wave >= N_TILES) return;                 // wave-uniform: EXEC stays all-ones
    const int lane = threadIdx.x & 31;
    const int half = lane >> 4;                  // 0: K pair {0,1}, 1: K pair {2,3}
    const int l16  = lane & 15;
    const size_t row = (size_t)wave * 16 + l16;  // A row for this lane
    const float* xr = x + row * D_IN + 2 * half;

    v8f c = {};
#pragma unroll
    for (int kk = 0; kk < D_IN; kk += 4) {
        v2f a, b;
        a.x = xr[kk];
        a.y = xr[kk + 1];
        const int kb = kk + 2 * half;
        b.x = W1[kb * D_HID + l16];
        b.y = W1[(kb + 1) * D_HID + l16];
        c = __builtin_amdgcn_wmma_f32_16x16x4_f32(false, a, false, b,
                                                  (short)0, c, false, false);
    }
#pragma unroll
    for (int j = 0; j < 8; ++j) {
        const size_t m = (size_t)wave * 16 + j + 8 * half;
        h1[m * D_HID + l16] = c[j];
    }
}

// ---------------- layer 1 aggregation ----------------

// out1[n][c] = b1[c] + dinv[n]^2 * h1[n][c]   (bias + self-loop term)
__global__ void k_init_out1(const float* __restrict__ h1,
                            const float* __restrict__ dinv,
                            const float* __restrict__ b1,
                            float* __restrict__ out1) {
    int t = blockIdx.x * blockDim.x + threadIdx.x;
    if (t >= N_NODES * D_HID) return;
    int n = t >> 4, c = t & 15;
    float di = dinv[n];
    out1[t] = b1[c] + di * di * h1[t];
}

// out1[d][c] += dinv[s]*dinv[d]*h1[s][c] over edges; 16 lanes per edge
__global__ void k_scatter1(const int* __restrict__ src,
                           const int* __restrict__ dst,
                           const float* __restrict__ h1,
                           const float* __restrict__ dinv,
                           float* __restrict__ out1) {
    long long t = (long long)blockIdx.x * blockDim.x + threadIdx.x;
    if (t >= (long long)N_EDGES * 16) return;
    int e = (int)(t >> 4), c = (int)(t & 15);
    int s = src[e], d = dst[e];
    float w = dinv[s] * dinv[d];
    atomicAdd(&out1[(size_t)d * D_HID + c], w * h1[(size_t)s * D_HID + c]);
}

// ---------------- layer 2 GEMM: h2 = relu(out1) @ W2 (WMMA, N padded to 16) ----------------

__global__ void k_gemm2(const float* __restrict__ o1,
                        const float* __restrict__ W2,
                        float* __restrict__ h2) {
    const int wave = (blockIdx.x * blockDim.x + threadIdx.x) >> 5;
    if (wave >= N_TILES) return;
    const int lane = threadIdx.x & 31;
    const int half = lane >> 4;
    const int l16  = lane & 15;
    const size_t row = (size_t)wave * 16 + l16;
    const float* ar = o1 + row * D_HID + 2 * half;
    const bool colOK = (l16 < N_CLS);

    v8f c = {};
#pragma unroll
    for (int kk = 0; kk < D_HID; kk += 4) {
        v2f a, b;
        a.x = fmaxf(ar[kk], 0.0f);       // ReLU folded into A read
        a.y = fmaxf(ar[kk + 1], 0.0f);
        const int kb = kk + 2 * half;
        b.x = colOK ? W2[kb * N_CLS + l16] : 0.0f;
        b.y = colOK ? W2[(kb + 1) * N_CLS + l16] : 0.0f;
        c = __builtin_amdgcn_wmma_f32_16x16x4_f32(false, a, false, b,
                                                  (short)0, c, false, false);
    }
#pragma unroll
    for (int j = 0; j < 8; ++j) {
        const size_t m = (size_t)wave * 16 + j + 8 * half;
        h2[m * 16 + l16] = c[j];         // padded stride 16; cols >= 10 are zero
    }
}

// ---------------- layer 2 aggregation (stride-16 padded accumulator) ----------------

__global__ void k_init_out2(const float* __restrict__ h2,
                            const float* __restrict__ dinv,
                            const float* __restrict__ b2,
                            float* __restrict__ out2) {
    int t = blockIdx.x * blockDim.x + threadIdx.x;
    if (t >= N_NODES * 16) return;
    int n = t >> 4, c = t & 15;
    if (c >= N_CLS) return;
    float di = dinv[n];
    out2[t] = b2[c] + di * di * h2[t];
}

__global__ void k_scatter2(const int* __restrict__ src,
                           const int* __restrict__ dst,
                           const float* __restrict__ h2,
                           const float* __restrict__ dinv,
                           float* __restrict__ out2) {
    long long t = (long long)blockIdx.x * blockDim.x + threadIdx.x;
    if (t >= (long long)N_EDGES * 16) return;
    int e = (int)(t >> 4), c = (int)(t & 15);
    if (c >= N_CLS) return;
    int s = src[e], d = dst[e];
    float w = dinv[s] * dinv[d];
    atomicAdd(&out2[(size_t)d * 16 + c], w * h2[(size_t)s * 16 + c]);
}

// ---------------- log-softmax over 10 classes ----------------

__global__ void k_logsoftmax(const float* __restrict__ out2, float* __restrict__ y) {
    int n = blockIdx.x * blockDim.x + threadIdx.x;
    if (n >= N_NODES) return;
    float v[N_CLS];
    float m = -INFINITY;
#pragma unroll
    for (int c = 0; c < N_CLS; ++c) { v[c] = out2[(size_t)n * 16 + c]; m = fmaxf(m, v[c]); }
    float s = 0.0f;
#pragma unroll
    for (int c = 0; c < N_CLS; ++c) s += __expf(v[c] - m);
    float lse = m + __logf(s);
#pragma unroll
    for (int c = 0; c < N_CLS; ++c) y[(size_t)n * N_CLS + c] = v[c] - lse;
}

// ---------------- host launch ----------------

extern "C" void kernel_launch(void* const* d_in, const int* in_sizes, int n_in,
                              void* d_out, int out_size, void* d_ws, size_t ws_size,
                              hipStream_t stream) {
    const float* x  = (const float*)d_in[0];
    const int*   ei = (const int*)  d_in[1];   // [2, E]
    const float* W1 = (const float*)d_in[2];
    const float* b1 = (const float*)d_in[3];
    const float* W2 = (const float*)d_in[4];
    const float* b2 = (const float*)d_in[5];
    const int* src = ei;
    const int* dst = ei + N_EDGES;

    float* ws   = (float*)d_ws;
    float* dinv = ws;                                   // N
    float* h1   = dinv + N_NODES;                       // N*16
    float* o1   = h1 + (size_t)N_NODES * 16;            // N*16
    float* h2   = o1 + (size_t)N_NODES * 16;            // N*16 (padded)
    float* o2   = h2 + (size_t)N_NODES * 16;            // N*16 (padded)
    float* y    = (float*)d_out;                        // N*10

    const int B = 256;
    const int gN   = (N_NODES + B - 1) / B;
    const int gE   = (N_EDGES + B - 1) / B;
    const int gN16 = (N_NODES * 16 + B - 1) / B;
    const long long e16 = (long long)N_EDGES * 16;
    const int gE16 = (int)((e16 + B - 1) / B);
    const int gW   = (N_TILES * 32 + B - 1) / B;        // 1 wave per 16-row tile

    k_init_deg   <<<gN,   B, 0, stream>>>(dinv);
    k_count_deg  <<<gE,   B, 0, stream>>>(dst, dinv);
    k_deg_to_dinv<<<gN,   B, 0, stream>>>(dinv);

    k_gemm1      <<<gW,   B, 0, stream>>>(x, W1, h1);
    k_init_out1  <<<gN16, B, 0, stream>>>(h1, dinv, b1, o1);
    k_scatter1   <<<gE16, B, 0, stream>>>(src, dst, h1, dinv, o1);

    k_gemm2      <<<gW,   B, 0, stream>>>(o1, W2, h2);
    k_init_out2  <<<gN16, B, 0, stream>>>(h2, dinv, b2, o2);
    k_scatter2   <<<gE16, B, 0, stream>>>(src, dst, h2, dinv, o2);

    k_logsoftmax <<<gN,   B, 0, stream>>>(o2, y);
}